// StripePolynomial2d_16071767621865
// MI455X (gfx1250) — compile-verified
//
#include <hip/hip_runtime.h>
#include <stdint.h>

typedef __attribute__((ext_vector_type(4))) unsigned int v4u;
typedef __attribute__((ext_vector_type(8))) unsigned int v8u;

#define PIX   (512 * 512)      // 262144 pixels
#define NWND  129              // weights per (stripe,o,c)
#define NGRP  36               // 4 stripes * 3 o * 3 c
#define NRAW  (NGRP * NWND)    // 4644 floats = 18576 bytes

// Transformed table: per group, 64 segments, float4 {A0,A1,A2,pad}
// value(seg, t) = A0 + t*A1 + t^2*A2,  t = tn - seg in [0,1)

__global__ __launch_bounds__(256) void stripe_poly_kernel(
    const float* __restrict__ x, const float* __restrict__ Wg,
    float* __restrict__ out)
{
    __shared__ float  sRaw[NRAW];
    __shared__ float4 sT[NGRP * 64];

    // ---- Stage 1: TDM async DMA of the raw weight table into LDS ----
    // One wave per block issues the tensor load (EXEC ignored by TDM).
    if (threadIdx.x < 32u) {
        uint64_t ga  = (uint64_t)(uintptr_t)Wg;
        uint32_t lds = (uint32_t)(uintptr_t)&sRaw[0];   // low 32 bits of generic ptr = LDS offset
        // D# group 0: count=1 | lds_addr | global_addr(57b) | type=2
        v4u g0 = { 1u,
                   lds,
                   (uint32_t)ga,
                   (((uint32_t)(ga >> 32)) & 0x01FFFFFFu) | 0x80000000u };
        // D# group 1: data_size=4B; tensor_dim0=4644; tensor_dim1=1;
        //             tile_dim0=4644; tile_dim1=1; stride0=4644
        v8u g1 = { 0x00020000u,     // workgroup_mask=0, data_size=2 (4B)
                   0x12240000u,     // tensor_dim0[15:0]=4644 in bits[63:48]
                   0x00010000u,     // tensor_dim0 hi=0, tensor_dim1 lo=1
                   0x12240000u,     // tensor_dim1 hi=0, tile_dim0=4644
                   0x00000001u,     // tile_dim1=1, tile_dim2=0
                   4644u,           // tensor_dim0_stride lo32
                   0u, 0u };
        asm volatile("tensor_load_to_lds %0, %1" :: "s"(g0), "s"(g1) : "memory");
        __builtin_amdgcn_s_wait_tensorcnt(0);
    }
    __syncthreads();

    // ---- Stage 2: node-basis -> local monomial coefficients (padded f4) ----
    // xl = 2t-1;  L0=(xl^2-xl)/2, L1=1-xl^2, L2=(xl^2+xl)/2
    for (int idx = threadIdx.x; idx < NGRP * 64; idx += 256) {
        int s = idx & 63, g = idx >> 6;
        const float* wp = &sRaw[g * NWND + 2 * s];
        float w0 = wp[0], w1 = wp[1], w2 = wp[2];
        float q2 = 0.5f * (w0 + w2) - w1;   // coeff of xl^2
        float q1 = 0.5f * (w2 - w0);        // coeff of xl
        float A2 = 4.0f * q2;
        float A1 = 2.0f * q1 - 4.0f * q2;
        float A0 = q2 - q1 + w1;
        sT[idx] = make_float4(A0, A1, A2, 0.0f);
    }
    __syncthreads();

    // ---- Stage 3: main compute. thread = 1 pixel, all 8 batches, 3 outputs ----
    const int p   = blockIdx.x * 256 + threadIdx.x;   // pixel index, coalesced
    const int wco = p >> 9;
    const int hco = p & 511;

    float xs[8][3];
#pragma unroll
    for (int b = 0; b < 8; ++b)
#pragma unroll
        for (int c = 0; c < 3; ++c)
            xs[b][c] = x[((b * 3 + c) << 18) + p] * 0.125f;   // x/8 (pre-scaled)

    float acc[8][3] = {};

    const float K01 = 512.0f * (512.0f / 513.0f) / 511.0f;    // stripes 0,1: pos = w*K01 - 256
    const float K23 = 512.0f * (512.0f / 513.0f) / 1022.0f;   // stripes 2,3

#pragma unroll
    for (int st = 0; st < 4; ++st) {
        float pos;
        if (st <= 1)      pos = (float)wco * K01 - 256.0f;
        else if (st == 2) pos = (float)(wco + hco) * K23 - 256.0f;
        else              pos = (float)(wco - hco + 511) * K23 - 256.0f;

        // tn = (x+pos)/8 + 32 ; base = uniform-over-b part
        float base = pos * 0.125f + 32.0f;     // in [0, 63.88) -> segA in [0,63]
        float fA   = floorf(base);
        int   segA = (int)fA;
        float r    = base - fA;                // u = r + x/8 in [0, 1.125)
        bool  canB = (segA < 63);
        int   segB = canB ? segA + 1 : segA;   // clamped like the reference

#pragma unroll
        for (int c = 0; c < 3; ++c) {
            float4 pa[3], pb[3];
#pragma unroll
            for (int o = 0; o < 3; ++o) {
                int g = (st * 3 + o) * 3 + c;
                pa[o] = sT[g * 64 + segA];     // ds_load_b128, serves all 8 batches
                pb[o] = sT[g * 64 + segB];
            }
#pragma unroll
            for (int b = 0; b < 8; ++b) {
                float u   = r + xs[b][c];
                bool  sel = canB && (u >= 1.0f);       // seg = segA+1 for this batch
                float t   = sel ? (u - 1.0f) : u;      // local coord (extrapolates at seg 63)
#pragma unroll
                for (int o = 0; o < 3; ++o) {
                    float c0 = sel ? pb[o].x : pa[o].x;
                    float c1 = sel ? pb[o].y : pa[o].y;
                    float c2 = sel ? pb[o].z : pa[o].z;
                    acc[b][o] += fmaf(t, fmaf(t, c2, c1), c0);
                }
            }
        }
    }

#pragma unroll
    for (int b = 0; b < 8; ++b)
#pragma unroll
        for (int o = 0; o < 3; ++o)
            out[((b * 3 + o) << 18) + p] = acc[b][o] * 0.25f;   // /(2*ROT)
}

extern "C" void kernel_launch(void* const* d_in, const int* in_sizes, int n_in,
                              void* d_out, int out_size, void* d_ws, size_t ws_size,
                              hipStream_t stream) {
    (void)in_sizes; (void)n_in; (void)d_ws; (void)ws_size; (void)out_size;
    const float* x  = (const float*)d_in[0];
    const float* Wg = (const float*)d_in[1];
    float* out = (float*)d_out;
    stripe_poly_kernel<<<dim3(PIX / 256), dim3(256), 0, stream>>>(x, Wg, out);
}